// DNFLayer_43344809952120
// MI455X (gfx1250) — compile-verified
//
#include <hip/hip_runtime.h>
#include <hip/hip_bf16.h>

// ---------------------------------------------------------------------------
// DNF layer for MI455X (gfx1250, wave32).
// B=32, N=32, P=992, P0=16, P1=32, P2=16, R=3, D=8, num_in=112.
// No matmul structure exists (product of affine forms), so no WMMA; the
// CDNA5-specific path used is the Tensor Data Mover: each block DMAs the
// 5400-float coefficient table (w, c, ok) from global into LDS with
// tensor_load_to_lds and synchronizes with s_wait_tensorcnt.
// ---------------------------------------------------------------------------

#define B_     32
#define N_     32
#define PPERM  992          // N*(N-1)
#define NIN    112
#define RD     24           // R*D
#define COEF_N 5400         // 2688 w + 2688 c + 24 ok
#define OUTROW 16912        // 16 + 1024 + 15872

typedef __attribute__((ext_vector_type(4))) unsigned int v4u;
typedef __attribute__((ext_vector_type(8))) int          v8i;
typedef __attribute__((ext_vector_type(4))) int          v4i;

// ---------------------------------------------------------------------------
// Kernel A: softmax of and_kernel -> (w, c), sigmoid of or_kernel -> ok.
// coef layout: [0,2688) w[rd*112+k], [2688,5376) c[rd*112+k], [5376,5400) ok[rd]
// ---------------------------------------------------------------------------
__global__ __launch_bounds__(256) void dnf_prep(const float* __restrict__ andk,
                                                const float* __restrict__ ork,
                                                float* __restrict__ coef) {
    int idx = blockIdx.x * 256 + threadIdx.x;
    if (idx < RD * NIN) {
        // and_kernel is (R, D, num_in, 3) row-major; idx == rd*112 + k.
        float a0 = andk[idx * 3 + 0];
        float a1 = andk[idx * 3 + 1];
        float a2 = andk[idx * 3 + 2];
        float m  = fmaxf(a0, fmaxf(a1, a2));
        float e0 = expf(a0 - m), e1 = expf(a1 - m), e2 = expf(a2 - m);
        float inv = 1.0f / (e0 + e1 + e2);
        float s0 = e0 * inv, s1 = e1 * inv, s2 = e2 * inv;
        coef[idx]            = s0 - s1;   // w
        coef[RD * NIN + idx] = s1 + s2;   // c
    }
    if (idx < RD) {
        coef[2 * RD * NIN + idx] = 1.0f / (1.0f + expf(-ork[idx]));  // ok
    }
}

// ---------------------------------------------------------------------------
// TDM: 1-row tile of `nelem` 4-byte elements from global -> LDS `lds_off`.
// D# per CDNA5 ISA ch.8: group0 = {flags, lds_addr, gaddr_lo, gaddr_hi|type},
// group1 = {data_size, tensor dims/strides, tile dims}. Groups 2/3 zero (2D).
// Toolchain builtin is the 6-arg form: (g0, g1, g2, g3, extra, cpol).
// ---------------------------------------------------------------------------
__device__ __forceinline__ void tdm_load_1d(const void* gsrc, unsigned lds_off,
                                            unsigned nelem) {
#if __has_builtin(__builtin_amdgcn_tensor_load_to_lds)
    unsigned long long ga = (unsigned long long)gsrc;
    v4u g0;
    g0.x = 1u;                                              // count=1, user D#
    g0.y = lds_off;                                         // lds_addr (bytes)
    g0.z = (unsigned)(ga & 0xFFFFFFFFu);                    // global_addr[31:0]
    g0.w = (unsigned)((ga >> 32) & 0x01FFFFFFu) | (2u << 30); // [56:32] + type=2
    v8i g1;
    g1[0] = (int)(2u << 16);                        // mask=0, data_size=2 (4B)
    g1[1] = (int)((nelem & 0xFFFFu) << 16);         // tensor_dim0[15:0]
    g1[2] = (int)(((nelem >> 16) & 0xFFFFu) | (1u << 16)); // td0[31:16], td1=1
    g1[3] = (int)((nelem & 0xFFFFu) << 16);         // tensor_dim1 hi=0, tile_dim0
    g1[4] = 1;                                      // tile_dim1=1, tile_dim2=0
    g1[5] = (int)nelem;                             // tensor_dim0_stride[31:0]
    g1[6] = (int)((nelem & 0xFFFFu) << 16);         // td0s hi=0, td1_stride[15:0]
    g1[7] = (int)(nelem >> 16);                     // td1_stride[47:16]
    v4i z4 = {0, 0, 0, 0};
    v8i z8 = {0, 0, 0, 0, 0, 0, 0, 0};
    __builtin_amdgcn_tensor_load_to_lds(g0, g1, z4, z4, z8, 0);
    __builtin_amdgcn_s_wait_tensorcnt(0);
#endif
}

// ---------------------------------------------------------------------------
// Kernel B: one (b,p) pair per lane.  Grid = 124 blocks * 256 thr = 31744.
// ---------------------------------------------------------------------------
__global__ __launch_bounds__(256) void dnf_main(const float* __restrict__ nul,
                                                const float* __restrict__ un,
                                                const float* __restrict__ bin,
                                                const float* __restrict__ coef,
                                                float* __restrict__ rule0,
                                                float* __restrict__ rule1,
                                                float* __restrict__ out) {
    __shared__ float lds_coef[COEF_N];

    // Low 32 bits of the flat address of an LDS object == its LDS byte offset
    // (see ISA 10.2).
    const unsigned lds_off = (unsigned)(uintptr_t)(void*)&lds_coef[0];

#if __has_builtin(__builtin_amdgcn_tensor_load_to_lds)
    if ((threadIdx.x >> 5) == 0) {       // one wave issues the DMA
        tdm_load_1d(coef, lds_off, COEF_N);
    }
    // Escape the LDS array (ptrtoint capture) + memory clobber: the TDM
    // builtin writes lds_coef behind the optimizer's back, so loads from it
    // must not be folded to undef or hoisted above this point.
    asm volatile("" : : "v"(lds_off) : "memory");
#else
    for (int t = threadIdx.x; t < COEF_N; t += 256) lds_coef[t] = coef[t];
#endif
    __syncthreads();

    const int gid = blockIdx.x * 256 + threadIdx.x;     // 0..31743
    const int b   = gid / PPERM;
    const int p   = gid - b * PPERM;
    const int i   = p / (N_ - 1);
    const int jr  = p - i * (N_ - 1);
    const int j   = jr + (jr >= i ? 1 : 0);
    const int col1 = (jr >= i) ? i : (i - 1);           // i - (i > j)

    // ---- gather x[112] into registers (all float4 aligned) ----
    float x[NIN];
    {
        const float4* s = (const float4*)(nul + b * 16);
        #pragma unroll
        for (int q = 0; q < 4; ++q) {
            float4 v = s[q];
            x[4*q+0] = v.x; x[4*q+1] = v.y; x[4*q+2] = v.z; x[4*q+3] = v.w;
        }
    }
    {
        const float4* s = (const float4*)(un + (b * N_ + i) * 32);
        #pragma unroll
        for (int q = 0; q < 8; ++q) {
            float4 v = s[q];
            x[16+4*q+0] = v.x; x[16+4*q+1] = v.y; x[16+4*q+2] = v.z; x[16+4*q+3] = v.w;
        }
    }
    {
        const float4* s = (const float4*)(un + (b * N_ + j) * 32);
        #pragma unroll
        for (int q = 0; q < 8; ++q) {
            float4 v = s[q];
            x[48+4*q+0] = v.x; x[48+4*q+1] = v.y; x[48+4*q+2] = v.z; x[48+4*q+3] = v.w;
        }
    }
    {
        const float4* s = (const float4*)(bin + ((b * N_ + i) * (N_ - 1) + jr) * 16);
        #pragma unroll
        for (int q = 0; q < 4; ++q) {
            float4 v = s[q];
            x[80+4*q+0] = v.x; x[80+4*q+1] = v.y; x[80+4*q+2] = v.z; x[80+4*q+3] = v.w;
        }
    }
    {
        const float4* s = (const float4*)(bin + ((b * N_ + j) * (N_ - 1) + col1) * 16);
        #pragma unroll
        for (int q = 0; q < 4; ++q) {
            float4 v = s[q];
            x[96+4*q+0] = v.x; x[96+4*q+1] = v.y; x[96+4*q+2] = v.z; x[96+4*q+3] = v.w;
        }
    }

    // ---- conjuncts / disjuncts ----
    float dis[3];
    #pragma unroll 1
    for (int r = 0; r < 3; ++r) {
        float dp = 1.0f;
        #pragma unroll 1
        for (int d = 0; d < 8; ++d) {
            const int rd = r * 8 + d;
            const float* wrow = &lds_coef[rd * NIN];            // LDS broadcast
            const float* crow = &lds_coef[RD * NIN + rd * NIN]; // LDS broadcast
            float p0 = 1.0f, p1 = 1.0f, p2 = 1.0f, p3 = 1.0f;  // split chains
            #pragma unroll
            for (int k = 0; k < NIN; k += 4) {
                p0 *= fmaf(x[k+0], wrow[k+0], crow[k+0]);
                p1 *= fmaf(x[k+1], wrow[k+1], crow[k+1]);
                p2 *= fmaf(x[k+2], wrow[k+2], crow[k+2]);
                p3 *= fmaf(x[k+3], wrow[k+3], crow[k+3]);
            }
            const float prod = (p0 * p1) * (p2 * p3);
            const float okv  = lds_coef[2 * RD * NIN + rd];
            dp *= (1.0f - prod * okv);
        }
        dis[r] = 1.0f - dp;
    }

    // ---- per-pair outputs ----
    rule0[gid] = dis[0];
    rule1[gid] = dis[1];

    // binary output row (b, i, jr): channels 0..14 copy, 15 merged with dis[2]
    float* ob = out + (size_t)b * OUTROW + 1040 + (size_t)p * 16;
    float4 o0 = {x[80], x[81], x[82], x[83]};
    float4 o1 = {x[84], x[85], x[86], x[87]};
    float4 o2 = {x[88], x[89], x[90], x[91]};
    float4 o3 = {x[92], x[93], x[94],
                 1.0f - (1.0f - x[95]) * (1.0f - dis[2])};
    ((float4*)ob)[0] = o0;
    ((float4*)ob)[1] = o1;
    ((float4*)ob)[2] = o2;
    ((float4*)ob)[3] = o3;
}

// ---------------------------------------------------------------------------
// Kernel C: nullary/unary probsum reductions + merged copies. 32 blocks x 64.
// ---------------------------------------------------------------------------
__global__ __launch_bounds__(64) void dnf_tail(const float* __restrict__ nul,
                                               const float* __restrict__ un,
                                               const float* __restrict__ rule0,
                                               const float* __restrict__ rule1,
                                               float* __restrict__ out) {
    const int b = blockIdx.x;
    const int t = threadIdx.x;
    if (t < 32) {
        const int i = t;
        float pu = 1.0f;                           // prod(1 - rule1) over jr
        for (int jr = 0; jr < N_ - 1; ++jr)
            pu *= (1.0f - rule1[b * PPERM + i * (N_ - 1) + jr]);
        const float* src = un + (b * N_ + i) * 32;
        float*       dst = out + (size_t)b * OUTROW + 16 + i * 32;
        #pragma unroll
        for (int c = 0; c < 31; ++c) dst[c] = src[c];
        // merged = 1 - (1-old)*(1-new), new = 1-pu  =>  1 - (1-old)*pu
        dst[31] = 1.0f - (1.0f - src[31]) * pu;
    } else if (t == 32) {
        float pn = 1.0f;                           // prod(1 - rule0) over all p
        for (int p = 0; p < PPERM; ++p)
            pn *= (1.0f - rule0[b * PPERM + p]);
        const float* src = nul + b * 16;
        float*       dst = out + (size_t)b * OUTROW;
        for (int c = 0; c < 15; ++c) dst[c] = src[c];
        dst[15] = 1.0f - (1.0f - src[15]) * pn;
    }
}

// ---------------------------------------------------------------------------
extern "C" void kernel_launch(void* const* d_in, const int* in_sizes, int n_in,
                              void* d_out, int out_size, void* d_ws, size_t ws_size,
                              hipStream_t stream) {
    const float* nul  = (const float*)d_in[0];  // (32,16)
    const float* un   = (const float*)d_in[1];  // (32,32,32)
    const float* bin  = (const float*)d_in[2];  // (32,32,31,16)
    const float* andk = (const float*)d_in[3];  // (3,8,112,3)
    const float* ork  = (const float*)d_in[4];  // (3,8)
    float* out = (float*)d_out;                 // (32,16912)

    float* ws   = (float*)d_ws;
    float* coef = ws;                           // 5400 floats (+8 pad)
    float* r0   = ws + 5408;                    // 31744 floats
    float* r1   = ws + 5408 + 31744;            // 31744 floats

    dnf_prep<<<11, 256, 0, stream>>>(andk, ork, coef);
    dnf_main<<<(B_ * PPERM) / 256, 256, 0, stream>>>(nul, un, bin, coef, r0, r1, out);
    dnf_tail<<<B_, 64, 0, stream>>>(nul, un, r0, r1, out);
}